// SheafEdgeDecoder_66864050864372
// MI455X (gfx1250) — compile-verified
//
#include <hip/hip_runtime.h>

// Per-lane fragment types for V_WMMA_F32_16X16X4_F32 (wave32):
//  A: 16x4 f32  -> 64 elems / 32 lanes = 2 VGPRs  (v2f)
//  B: 4x16 f32  -> 64 elems / 32 lanes = 2 VGPRs  (v2f)
//  C/D: 16x16 f32 -> 256 elems / 32 lanes = 8 VGPRs (v8f)
typedef __attribute__((ext_vector_type(2))) float v2f;
typedef __attribute__((ext_vector_type(8))) float v8f;

#define D_FEAT 128

// One wave handles 16 edges. For K-chunk k:
//   A[m][kk] = x[src[base+m]][k+kk]   (m = edge slot, rows of A)
//   B[kk][n] = x[dst[base+n]][k+kk]   (n = edge slot, cols of B)
// => D = sum_k A*B ;  D[e][e] = dot(x[src[e]], x[dst[e]])
//
// ISA layouts make A and B fragment loads identical per lane:
//   lane L (m = L&15, h = L>>4): float2 at row[m] + k + 2h.
__global__ __launch_bounds__(256) void sheaf_edge_dot_wmma(
    const float* __restrict__ x,
    const int*   __restrict__ edge_index,   // [2, E] flat: src = [e], dst = [E+e]
    float*       __restrict__ out,          // [E]
    int E)
{
    const int lane = threadIdx.x & 31;
    const int waveInBlk = threadIdx.x >> 5;
    const long long tile = (long long)blockIdx.x * (blockDim.x >> 5) + waveInBlk;
    const long long base = tile * 16;
    if (base >= E) return;                  // wave-uniform: EXEC stays all-ones past here

    const int m = lane & 15;                // edge slot within this wave's tile
    const int h = lane >> 4;                // K-half selector (K pairs {0,1} vs {2,3})

    long long e = base + m;
    if (e >= (long long)E) e = (long long)E - 1;   // tail clamp (VALU select, no EXEC change)

    const int srow = edge_index[e];
    const int drow = edge_index[(long long)E + e];

    const float* __restrict__ sp = x + (long long)srow * D_FEAT + 2 * h;
    const float* __restrict__ dp = x + (long long)drow * D_FEAT + 2 * h;

    v8f c = {0.f, 0.f, 0.f, 0.f, 0.f, 0.f, 0.f, 0.f};

#pragma unroll
    for (int k = 0; k < D_FEAT; k += 4) {
        v2f a = *(const v2f*)(sp + k);      // global_load_b64, 8B aligned
        v2f b = *(const v2f*)(dp + k);
        // args: (neg_a, A, neg_b, B, c_mod, C, reuse_a, reuse_b)
        c = __builtin_amdgcn_wmma_f32_16x16x4_f32(
                false, a, false, b, (short)0, c, false, false);
    }

    // Diagonal extraction per ISA C/D layout:
    //   e < 8 : lane e,     VGPR e      (M=r, N=lane)
    //   e >= 8: lane 24+e-8, VGPR e-8   (M=8+r, N=lane-16)
    float val = 0.f;
#pragma unroll
    for (int r = 0; r < 8; ++r) {
        val = (lane == r)      ? c[r] : val;
        val = (lane == 24 + r) ? c[r] : val;
    }

    const bool active = (lane < 8) || (lane >= 24);
    const long long eo = base + ((lane < 8) ? lane : (lane - 16));
    if (active && eo < (long long)E) out[eo] = val;
}

extern "C" void kernel_launch(void* const* d_in, const int* in_sizes, int n_in,
                              void* d_out, int out_size, void* d_ws, size_t ws_size,
                              hipStream_t stream) {
    const float* x   = (const float*)d_in[0];
    const int*   ei  = (const int*)d_in[1];     // integer input -> int32 per harness convention
    float*       out = (float*)d_out;

    const int E = in_sizes[1] / 2;              // edge_index is [2, E]
    const int block = 256;                      // 8 waves
    const int wavesPerBlock = block / 32;
    const int tiles = (E + 15) / 16;            // 16 edges per wave
    const int grid  = (tiles + wavesPerBlock - 1) / wavesPerBlock;

    hipLaunchKernelGGL(sheaf_edge_dot_wmma, dim3(grid), dim3(block), 0, stream,
                       x, ei, out, E);
}